// SelectiveSSM_22522808500606
// MI455X (gfx1250) — compile-verified
//
#include <hip/hip_runtime.h>

#define D_INNER 2048
#define D_STATE 16
#define DT_RANK 64
#define BATCH   2
#define SEQ     2048
#define KX      (DT_RANK + 2 * D_STATE)   // 96
#define M_TOTAL (BATCH * SEQ)             // 4096
#define NCH     (BATCH * D_INNER)         // 4096 scan channels
#define CHUNKS  32
#define CLEN    (SEQ / CHUNKS)            // 64

typedef __attribute__((ext_vector_type(2))) float v2f;
typedef __attribute__((ext_vector_type(4))) float v4f;
typedef __attribute__((ext_vector_type(8))) float v8f;

// D = A(16x4, f32) x B(4x16, f32) + C(16x16, f32) -- v_wmma_f32_16x16x4_f32
__device__ __forceinline__ v8f wmma_f32_k4(v2f a, v2f b, v8f c) {
    return __builtin_amdgcn_wmma_f32_16x16x4_f32(
        false, a, false, b, (short)0, c, false, false);
}

// Broadcast lane `l`'s value to all lanes via v_readlane_b32 (SGPR result,
// becomes a scalar operand of the consuming VALU op -- no LDS traffic).
__device__ __forceinline__ float bcast(float v, int l) {
    return __int_as_float(__builtin_amdgcn_readlane(__float_as_int(v), l));
}

// ---------------------------------------------------------------------------
// Kernel 1: x_dbl[m, n] = sum_d u[m, d] * W_x[n, d]
//   M = 4096, N = 96, K = 2048.  One wave per 16-row M tile computes ALL SIX
//   16-col N tiles (6 x v8f accumulators) so u is streamed exactly once.
// ---------------------------------------------------------------------------
__global__ void __launch_bounds__(64)
k_gemm_xdbl(const float* __restrict__ u, const float* __restrict__ Wx,
            float* __restrict__ xdbl)
{
    const int lane = threadIdx.x & 31;
    const int wave = threadIdx.x >> 5;
    const int mt   = blockIdx.x * 2 + wave;   // 0..255
    const int row  = lane & 15;
    const int kh   = lane >> 4;

    const float* aptr = u  + (size_t)(mt * 16 + row) * D_INNER + 2 * kh;
    const float* bptr = Wx + (size_t)row * D_INNER + 2 * kh;

    v8f acc[6] = {};
#pragma unroll 2
    for (int k = 0; k < D_INNER; k += 4) {
        v2f a = *(const v2f*)(aptr + k);
#pragma unroll
        for (int nt = 0; nt < 6; ++nt) {
            v2f b = *(const v2f*)(bptr + (size_t)nt * 16 * D_INNER + k);
            acc[nt] = wmma_f32_k4(a, b, acc[nt]);
        }
    }

#pragma unroll
    for (int nt = 0; nt < 6; ++nt) {
        float* o = xdbl + (size_t)(mt * 16 + 8 * kh) * KX + nt * 16 + row;
#pragma unroll
        for (int r = 0; r < 8; ++r) o[r * KX] = acc[nt][r];
    }
}

// ---------------------------------------------------------------------------
// Kernel 2: dt[m, d] = softplus( sum_r x_dbl[m, r] * W_dt[d, r] + b_dt[d] )
//   M = 4096, N = 2048, K = 64.  One wave per 16x16 tile.
// ---------------------------------------------------------------------------
__global__ void __launch_bounds__(128)
k_gemm_dt(const float* __restrict__ xdbl, const float* __restrict__ Wdt,
          const float* __restrict__ bdt, float* __restrict__ dt)
{
    const int lane = threadIdx.x & 31;
    const int wave = threadIdx.x >> 5;
    const int w    = blockIdx.x * 4 + wave;   // 0..32767
    const int nt   = w & 127;
    const int mt   = w >> 7;
    const int row  = lane & 15;
    const int kh   = lane >> 4;

    const float* aptr = xdbl + (size_t)(mt * 16 + row) * KX + 2 * kh;
    const float* bptr = Wdt  + (size_t)(nt * 16 + row) * DT_RANK + 2 * kh;

    v8f acc = {};
#pragma unroll
    for (int k = 0; k < DT_RANK; k += 4) {
        v2f a = *(const v2f*)(aptr + k);
        v2f b = *(const v2f*)(bptr + k);
        acc = wmma_f32_k4(a, b, acc);
    }

    const int   n    = nt * 16 + row;
    const float bias = bdt[n];
    float* o = dt + (size_t)(mt * 16 + 8 * kh) * D_INNER + n;
#pragma unroll
    for (int r = 0; r < 8; ++r) {
        float x = acc[r] + bias;
        o[r * D_INNER] = (x > 20.0f) ? x : log1pf(__expf(x));  // softplus
    }
}

// ---------------------------------------------------------------------------
// Chunked parallel scan: h_t = a_t*h_{t-1} + b_t is associative, so split L
// into 32 chunks of 64.  Pass 1: per-chunk (prod a, h_end from h=0).
// Pass 2: 32-step serial composition -> h_init per chunk.  Pass 3: redo the
// chunk recurrence from h_init and emit y.  4096 waves in passes 1/3.
// ---------------------------------------------------------------------------
__global__ void __launch_bounds__(256)
k_scan_pass1(const float* __restrict__ u, const float* __restrict__ A_log,
             const float* __restrict__ xdbl, const float* __restrict__ dt,
             float* __restrict__ hend, float* __restrict__ aprod)
{
    const int c    = blockIdx.x & (CHUNKS - 1);
    const int ch   = (blockIdx.x >> 5) * 256 + threadIdx.x;   // 0..4095
    const int b    = ch >> 11;
    const int d    = ch & (D_INNER - 1);
    const int lane = threadIdx.x & 31;

    float A[D_STATE], h[D_STATE], ap[D_STATE];
#pragma unroll
    for (int n = 0; n < D_STATE; ++n) {
        A[n]  = -__expf(A_log[d * D_STATE + n]);
        h[n]  = 0.0f;
        ap[n] = 1.0f;
    }

    const int t0 = c * CLEN;
    for (int t = t0; t < t0 + CLEN; ++t) {
        const int   m   = b * SEQ + t;
        const float bc  = xdbl[(size_t)m * KX + DT_RANK + lane];
        const float dtv = dt[(size_t)m * D_INNER + d];
        const float du  = dtv * u[(size_t)m * D_INNER + d];
#pragma unroll
        for (int n = 0; n < D_STATE; ++n) {
            const float a = __expf(dtv * A[n]);
            h[n]  = a * h[n] + du * bcast(bc, n);
            ap[n] *= a;
        }
    }

    const size_t base = ((size_t)c * NCH + ch) * D_STATE;
#pragma unroll
    for (int q = 0; q < 4; ++q) {
        *(v4f*)(hend  + base + 4 * q) = *(v4f*)(h  + 4 * q);
        *(v4f*)(aprod + base + 4 * q) = *(v4f*)(ap + 4 * q);
    }
}

__global__ void __launch_bounds__(256)
k_scan_pass2(const float* __restrict__ hend, const float* __restrict__ aprod,
             float* __restrict__ hinit)
{
    const int idx = blockIdx.x * 256 + threadIdx.x;   // 0..65535 = (ch, n)
    float h = 0.0f;
    for (int c = 0; c < CHUNKS; ++c) {
        const size_t off = (size_t)c * NCH * D_STATE + idx;
        hinit[off] = h;
        h = aprod[off] * h + hend[off];
    }
}

__global__ void __launch_bounds__(256)
k_scan_pass3(const float* __restrict__ u, const float* __restrict__ A_log,
             const float* __restrict__ Dvec, const float* __restrict__ xdbl,
             const float* __restrict__ dt, const float* __restrict__ hinit,
             float* __restrict__ y)
{
    const int c    = blockIdx.x & (CHUNKS - 1);
    const int ch   = (blockIdx.x >> 5) * 256 + threadIdx.x;
    const int b    = ch >> 11;
    const int d    = ch & (D_INNER - 1);
    const int lane = threadIdx.x & 31;

    float A[D_STATE], h[D_STATE];
#pragma unroll
    for (int n = 0; n < D_STATE; ++n)
        A[n] = -__expf(A_log[d * D_STATE + n]);
    const float Dd = Dvec[d];

    const size_t base = ((size_t)c * NCH + ch) * D_STATE;
#pragma unroll
    for (int q = 0; q < 4; ++q)
        *(v4f*)(h + 4 * q) = *(const v4f*)(hinit + base + 4 * q);

    const int t0 = c * CLEN;
    for (int t = t0; t < t0 + CLEN; ++t) {
        const int   m   = b * SEQ + t;
        const float bc  = xdbl[(size_t)m * KX + DT_RANK + lane];
        const float dtv = dt[(size_t)m * D_INNER + d];
        const float uv  = u[(size_t)m * D_INNER + d];
        const float du  = dtv * uv;

        float yt = 0.0f;
#pragma unroll
        for (int n = 0; n < D_STATE; ++n) {
            h[n] = __expf(dtv * A[n]) * h[n] + du * bcast(bc, n);
            yt  += h[n] * bcast(bc, n + 16);
        }
        y[(size_t)m * D_INNER + d] = yt + uv * Dd;
    }
}

// Fallback serial scan (used only if workspace is too small for the chunked
// path).  Same math, one 2048-step chain per channel.
__global__ void __launch_bounds__(256)
k_scan_serial(const float* __restrict__ u, const float* __restrict__ A_log,
              const float* __restrict__ Dvec, const float* __restrict__ xdbl,
              const float* __restrict__ dt, float* __restrict__ y)
{
    const int b    = blockIdx.x >> 3;
    const int d    = ((blockIdx.x & 7) << 8) + threadIdx.x;
    const int lane = threadIdx.x & 31;

    float A[D_STATE], h[D_STATE];
#pragma unroll
    for (int n = 0; n < D_STATE; ++n) {
        A[n] = -__expf(A_log[d * D_STATE + n]);
        h[n] = 0.0f;
    }
    const float Dd = Dvec[d];

    for (int t = 0; t < SEQ; ++t) {
        const int   m   = b * SEQ + t;
        const float bc  = xdbl[(size_t)m * KX + DT_RANK + lane];
        const float dtv = dt[(size_t)m * D_INNER + d];
        const float uv  = u[(size_t)m * D_INNER + d];
        const float du  = dtv * uv;

        float yt = 0.0f;
#pragma unroll
        for (int n = 0; n < D_STATE; ++n) {
            h[n] = __expf(dtv * A[n]) * h[n] + du * bcast(bc, n);
            yt  += h[n] * bcast(bc, n + 16);
        }
        y[(size_t)m * D_INNER + d] = yt + uv * Dd;
    }
}

// ---------------------------------------------------------------------------
extern "C" void kernel_launch(void* const* d_in, const int* in_sizes, int n_in,
                              void* d_out, int out_size, void* d_ws, size_t ws_size,
                              hipStream_t stream)
{
    const float* u     = (const float*)d_in[0];
    const float* A_log = (const float*)d_in[1];
    const float* Dv    = (const float*)d_in[2];
    const float* Wx    = (const float*)d_in[3];
    const float* Wdt   = (const float*)d_in[4];
    const float* bdt   = (const float*)d_in[5];
    float*       out   = (float*)d_out;

    const size_t n_xdbl = (size_t)M_TOTAL * KX;          //  393,216 f
    const size_t n_dt   = (size_t)M_TOTAL * D_INNER;     // 8,388,608 f
    const size_t n_st   = (size_t)CHUNKS * NCH * D_STATE;// 2,097,152 f

    float* xdbl  = (float*)d_ws;
    float* dt    = xdbl + n_xdbl;
    float* hend  = dt + n_dt;
    float* aprod = hend + n_st;
    float* hinit = aprod + n_st;

    const size_t need_chunked = (n_xdbl + n_dt + 3 * n_st) * sizeof(float);

    // GEMM1: 256 M-tiles, 2 waves/block
    k_gemm_xdbl<<<128, 64, 0, stream>>>(u, Wx, xdbl);
    // GEMM2: 32768 tiles, 4 waves/block
    k_gemm_dt<<<8192, 128, 0, stream>>>(xdbl, Wdt, bdt, dt);

    if (ws_size >= need_chunked) {
        // Chunked scan: (16 channel-groups x 32 chunks) blocks of 256
        k_scan_pass1<<<512, 256, 0, stream>>>(u, A_log, xdbl, dt, hend, aprod);
        k_scan_pass2<<<256, 256, 0, stream>>>(hend, aprod, hinit);
        k_scan_pass3<<<512, 256, 0, stream>>>(u, A_log, Dv, xdbl, dt, hinit, out);
    } else {
        k_scan_serial<<<16, 256, 0, stream>>>(u, A_log, Dv, xdbl, dt, out);
    }
}